// GatedCrossAttention_27324581937373
// MI455X (gfx1250) — compile-verified
//
#include <hip/hip_runtime.h>
#include <hip/hip_bf16.h>

#define BATCH 64
#define DIM 1024

typedef __attribute__((ext_vector_type(2))) float v2f;
typedef __attribute__((ext_vector_type(4))) float v4f;
typedef __attribute__((ext_vector_type(8))) float v8f;

// ---------------------------------------------------------------------------
// Y[M,N] = X[M,K] @ W[:, coff:coff+K].T + bias   (W row-major with leading dim
// ldw). One wave per 16x16 output tile, V_WMMA_F32_16X16X4_F32 over K.
//
// f32 WMMA lane layout (wave32):
//   A (16x4,  MxK): lane L -> row m = L&15, a[r] = A[m][2*(L>>4) + r]
//   B (4x16,  KxN): lane L -> col n = L&15, b[r] = B[2*(L>>4) + r][n]
//   C/D (16x16):    lane L -> n = L&15, c[r] = D[r + 8*(L>>4)][n]
// Both A and B operands are two consecutive-K floats of a row-major row.
// ---------------------------------------------------------------------------
__global__ __launch_bounds__(128) void gemm_nt_wmma_f32(
    const float* __restrict__ X, const float* __restrict__ W,
    const float* __restrict__ bias, float* __restrict__ Y,
    int M, int N, int K, int ldw, int coff) {
  const int wave = threadIdx.x >> 5;
  const int lane = threadIdx.x & 31;
  const int ntn  = N >> 4;                       // tiles along N
  const int tile = blockIdx.x * 4 + wave;
  if (tile >= (M >> 4) * ntn) return;            // wave-uniform; EXEC stays full
  const int trow = tile / ntn;
  const int tcol = tile - trow * ntn;
  const int half = lane >> 4;
  const int l15  = lane & 15;

  const float* xrow = X + (size_t)(trow * 16 + l15) * K + 2 * half;
  const float* wrow = W + (size_t)(tcol * 16 + l15) * ldw + coff + 2 * half;

  v8f acc = {0.f, 0.f, 0.f, 0.f, 0.f, 0.f, 0.f, 0.f};
#pragma unroll 8
  for (int k0 = 0; k0 < K; k0 += 4) {
    v2f a = *(const v2f*)(xrow + k0);
    v2f b = *(const v2f*)(wrow + k0);
    acc = __builtin_amdgcn_wmma_f32_16x16x4_f32(
        /*neg_a=*/false, a, /*neg_b=*/false, b,
        /*c_mod=*/(short)0, acc, /*reuse_a=*/false, /*reuse_b=*/false);
  }

  const float bv = bias[tcol * 16 + l15];
#pragma unroll
  for (int r = 0; r < 8; ++r) {
    const int m = r + 8 * half;
    Y[(size_t)(trow * 16 + m) * N + tcol * 16 + l15] = acc[r] + bv;
  }
}

// ---------------------------------------------------------------------------
// w1s[j] = sum_{d<K} Wg[j*ldw + d].  One wave per row, coalesced, wave-reduce.
// ---------------------------------------------------------------------------
__global__ __launch_bounds__(256) void rowsum_kernel(
    const float* __restrict__ Wg, float* __restrict__ w1s, int K, int ldw) {
  const int wave = threadIdx.x >> 5;
  const int lane = threadIdx.x & 31;
  const int row  = blockIdx.x * 8 + wave;
  const float* p = Wg + (size_t)row * ldw;
  float s = 0.f;
  for (int d = lane; d < K; d += 32) s += p[d];
#pragma unroll
  for (int off = 16; off >= 1; off >>= 1) s += __shfl_xor(s, off, 32);
  if (lane == 0) w1s[row] = s;
}

__device__ __forceinline__ float dsig2(float x) {
  // sigmoid(sigmoid(x))
  float s1 = __builtin_amdgcn_rcpf(1.0f + __expf(-x));
  return __builtin_amdgcn_rcpf(1.0f + __expf(-s1));
}

// ---------------------------------------------------------------------------
// out[b,i,j] = softmax_j( qp[b,i]*kp[b,j] * dsig2(qp[b,i]*w1s[j] + kg[b,j]) )
// One 256-thread block handles 4 consecutive rows (same batch b); the kp/kg/
// w1s row vectors are loaded once into registers and reused.  4 j's / thread.
// Streaming 256MB output -> non-temporal b128 stores.
// ---------------------------------------------------------------------------
__global__ __launch_bounds__(256) void attn_softmax_kernel(
    const float* __restrict__ qp, const float* __restrict__ kp,
    const float* __restrict__ kg, const float* __restrict__ w1s,
    float* __restrict__ out) {
  __shared__ float red_max[8];
  __shared__ float red_sum[8];
  const int t    = threadIdx.x;
  const int lane = t & 31;
  const int wave = t >> 5;
  const int row0 = blockIdx.x * 4;        // flattened (b,i); 4 | DIM so b fixed
  const int b    = row0 >> 10;

  const v4f kpv = ((const v4f*)(kp + (size_t)b * DIM))[t];
  const v4f kgv = ((const v4f*)(kg + (size_t)b * DIM))[t];
  const v4f wsv = ((const v4f*)w1s)[t];

  for (int r = 0; r < 4; ++r) {
    const int row  = row0 + r;
    const float qv = qp[row];             // qp is [b][i] flat == row index

    float v0 = qv * kpv.x * dsig2(qv * wsv.x + kgv.x);
    float v1 = qv * kpv.y * dsig2(qv * wsv.y + kgv.y);
    float v2 = qv * kpv.z * dsig2(qv * wsv.z + kgv.z);
    float v3 = qv * kpv.w * dsig2(qv * wsv.w + kgv.w);

    // row max
    float m = fmaxf(fmaxf(v0, v1), fmaxf(v2, v3));
#pragma unroll
    for (int off = 16; off >= 1; off >>= 1) m = fmaxf(m, __shfl_xor(m, off, 32));
    if (lane == 0) red_max[wave] = m;
    __syncthreads();
    float rm = red_max[0];
#pragma unroll
    for (int w = 1; w < 8; ++w) rm = fmaxf(rm, red_max[w]);

    // exp + row sum
    float e0 = __expf(v0 - rm), e1 = __expf(v1 - rm);
    float e2 = __expf(v2 - rm), e3 = __expf(v3 - rm);
    float s = (e0 + e1) + (e2 + e3);
#pragma unroll
    for (int off = 16; off >= 1; off >>= 1) s += __shfl_xor(s, off, 32);
    if (lane == 0) red_sum[wave] = s;
    __syncthreads();
    float tot = red_sum[0];
#pragma unroll
    for (int w = 1; w < 8; ++w) tot += red_sum[w];
    const float rinv = __builtin_amdgcn_rcpf(tot);

    v4f o = {e0 * rinv, e1 * rinv, e2 * rinv, e3 * rinv};
    __builtin_nontemporal_store(o, (v4f*)(out + (size_t)row * DIM) + t);
    __syncthreads();                      // red_* reuse next iteration
  }
}

extern "C" void kernel_launch(void* const* d_in, const int* in_sizes, int n_in,
                              void* d_out, int out_size, void* d_ws, size_t ws_size,
                              hipStream_t stream) {
  const float* q  = (const float*)d_in[0];
  const float* k  = (const float*)d_in[1];
  // d_in[2] = v : unused by the reference output
  const float* Wq = (const float*)d_in[3];
  const float* bq = (const float*)d_in[4];
  const float* Wk = (const float*)d_in[5];
  const float* bk = (const float*)d_in[6];
  const float* Wg = (const float*)d_in[7];
  const float* bg = (const float*)d_in[8];
  float* out = (float*)d_out;

  float* qp  = (float*)d_ws;            // 64*1024
  float* kp  = qp + BATCH * DIM;        // 64*1024
  float* kg  = kp + BATCH * DIM;        // 64*1024
  float* w1s = kg + BATCH * DIM;        // 1024

  const int gemm_grid = (BATCH / 16) * (DIM / 16) / 4;   // 64 blocks, 4 waves ea

  // q_proj and k_proj (independent)
  gemm_nt_wmma_f32<<<gemm_grid, 128, 0, stream>>>(q, Wq, bq, qp, BATCH, DIM, DIM, DIM, 0);
  gemm_nt_wmma_f32<<<gemm_grid, 128, 0, stream>>>(k, Wk, bk, kp, BATCH, DIM, DIM, DIM, 0);
  // w1 row sums (independent)
  rowsum_kernel<<<DIM / 8, 256, 0, stream>>>(Wg, w1s, DIM, 2 * DIM);
  // kg = k_proj @ W2.T + bg  (depends on kp; W2 = Wg[:, DIM:2*DIM])
  gemm_nt_wmma_f32<<<gemm_grid, 128, 0, stream>>>(kp, Wg, bg, kg, BATCH, DIM, DIM, 2 * DIM, DIM);
  // fused gate + softmax + streaming store
  attn_softmax_kernel<<<(BATCH * DIM) / 4, 256, 0, stream>>>(qp, kp, kg, w1s, out);
}